// GraphAttentionConv_7645041787181
// MI455X (gfx1250) — compile-verified
//
#include <hip/hip_runtime.h>
#include <math.h>

#define IN_DIM   128
#define OUT_DIM  128
#define NUM_HEAD 8
#define D_HEAD   16
#define NEG_SLOPE 0.2f
#define EPS 1e-10f

typedef __attribute__((ext_vector_type(2))) float v2f;
typedef __attribute__((ext_vector_type(4))) float v4f;
typedef __attribute__((ext_vector_type(8))) float v8f;

// float atomic max via signed/unsigned integer ordering trick (init to -inf bits)
__device__ __forceinline__ void atomic_max_f32(float* addr, float val) {
  if (val >= 0.0f) {
    atomicMax((int*)addr, __float_as_int(val));
  } else {
    atomicMin((unsigned int*)addr, __float_as_uint(val));
  }
}

// ---------------------------------------------------------------------------
// K0: init accumulators
// ---------------------------------------------------------------------------
__global__ void init_kernel(float* __restrict__ out, float* __restrict__ seg_max,
                            float* __restrict__ norm_sum, float* __restrict__ counts,
                            int Nn) {
  int total = Nn * OUT_DIM;
  for (int i = blockIdx.x * blockDim.x + threadIdx.x; i < total;
       i += gridDim.x * blockDim.x) {
    out[i] = 0.0f;
    if (i < Nn * NUM_HEAD) {
      seg_max[i]  = -INFINITY;
      norm_sum[i] = 0.0f;
    }
    if (i < Nn) counts[i] = 0.0f;
  }
}

// ---------------------------------------------------------------------------
// K1: hidden = x @ W + b  via V_WMMA_F32_16X16X4_F32 (full-precision fp32 WMMA)
// One wave computes a 16x16 tile; block of 8 waves covers a full row of 8
// column tiles (OUT_DIM=128). K-loop: 32 WMMAs of K=4.
//
// f32 16x16x4 operand layout (ISA 7.12.2):
//   A (16x4): lanes 0-15 -> M=lane, VGPR{0,1}=K{0,1}; lanes 16-31 -> K{2,3}
//   B (4x16): lanes 0-15 -> N=lane, VGPR{0,1}=K{0,1}; lanes 16-31 -> K{2,3}
//   C/D (16x16): VGPR r, lane<16 -> (M=r,   N=lane),
//                        lane>=16 -> (M=r+8, N=lane-16)
// ---------------------------------------------------------------------------
__global__ void gemm_wmma_kernel(const float* __restrict__ x,
                                 const float* __restrict__ W,
                                 const float* __restrict__ bvec,
                                 float* __restrict__ hidden) {
  const int wave = threadIdx.x >> 5;
  const int lane = threadIdx.x & 31;
  const int m0 = blockIdx.x * 16;
  const int n0 = wave * 16;
  const int l15  = lane & 15;        // A row / B col
  const int koff = (lane >> 4) << 1; // 0 for lanes 0-15, 2 for lanes 16-31

  v8f c = {0.f, 0.f, 0.f, 0.f, 0.f, 0.f, 0.f, 0.f};

  const float* ap = x + (size_t)(m0 + l15) * IN_DIM + koff;   // contiguous pair
  const float* bp = W + (size_t)koff * OUT_DIM + n0 + l15;    // stride OUT_DIM

#pragma unroll 4
  for (int k0 = 0; k0 < IN_DIM; k0 += 4) {
    v2f a = *(const v2f*)(ap + k0);
    v2f b;
    b.x = bp[(size_t)k0 * OUT_DIM];
    b.y = bp[(size_t)(k0 + 1) * OUT_DIM];
    c = __builtin_amdgcn_wmma_f32_16x16x4_f32(false, a, false, b,
                                              (short)0, c, false, false);
  }

  const float bias = bvec[n0 + l15];
  const int mhalf = (lane >> 4) << 3; // 0 or 8
#pragma unroll
  for (int r = 0; r < 8; ++r) {
    hidden[(size_t)(m0 + r + mhalf) * OUT_DIM + n0 + l15] = c[r] + bias;
  }
}

// ---------------------------------------------------------------------------
// K2: per-node attention coefficients
//   alpha_in[n,h]  = sum_j query[h,2j]   * hidden[n, 16h+j]
//   alpha_out[n,h] = sum_j query[h,2j+1] * hidden[n, 16h+j]
// (decomposition of the interleaved stack/reshape einsum in the reference)
// ---------------------------------------------------------------------------
__global__ void alpha_kernel(const float* __restrict__ hidden,
                             const float* __restrict__ query,
                             float* __restrict__ alpha_in,
                             float* __restrict__ alpha_out, int Nn) {
  int t = blockIdx.x * blockDim.x + threadIdx.x;
  if (t >= Nn * NUM_HEAD) return;
  const int n = t >> 3;
  const int h = t & 7;
  const float* hp = hidden + (size_t)n * OUT_DIM + h * D_HEAD;
  const float* qp = query + h * 2 * D_HEAD;
  float ai = 0.f, ao = 0.f;
#pragma unroll
  for (int j = 0; j < D_HEAD; ++j) {
    float hv = hp[j];
    ai += qp[2 * j] * hv;
    ao += qp[2 * j + 1] * hv;
  }
  alpha_in[t]  = ai;
  alpha_out[t] = ao;
}

// ---------------------------------------------------------------------------
// K3: per (edge, head): w = leaky_relu(a_in[src]+a_out[dst]); seg_max, counts
// ---------------------------------------------------------------------------
__global__ void attn_max_kernel(const int* __restrict__ edge_list,
                                const float* __restrict__ alpha_in,
                                const float* __restrict__ alpha_out,
                                float* __restrict__ seg_max,
                                float* __restrict__ counts, int E, int Nn) {
  long long t = (long long)blockIdx.x * blockDim.x + threadIdx.x;
  long long total = (long long)(E + Nn) * NUM_HEAD;
  if (t >= total) return;
  const int e = (int)(t >> 3);
  const int h = (int)(t & 7);
  int src, dst;
  if (e < E) { src = edge_list[2 * e]; dst = edge_list[2 * e + 1]; }
  else       { src = dst = e - E; }                          // self-loop
  float w = alpha_in[src * NUM_HEAD + h] + alpha_out[dst * NUM_HEAD + h];
  w = (w > 0.f) ? w : NEG_SLOPE * w;
  atomic_max_f32(&seg_max[dst * NUM_HEAD + h], w);
  if (h == 0) atomicAdd(&counts[dst], 1.0f);
}

// ---------------------------------------------------------------------------
// K4: one wave per edge. Recompute s = exp(w - max[dst])*ew, then
//   norm_sum[dst,h] += s  (8 lanes)
//   out[dst, d]     += s * hidden[src, d]  (each lane: 4 dims via float4 load)
// hidden (51 MB) is L2-resident -> random gathers + atomics stay in L2.
// ---------------------------------------------------------------------------
__global__ void scatter_kernel(const int* __restrict__ edge_list,
                               const float* __restrict__ edge_weight,
                               const float* __restrict__ hidden,
                               const float* __restrict__ alpha_in,
                               const float* __restrict__ alpha_out,
                               const float* __restrict__ seg_max,
                               float* __restrict__ norm_sum,
                               float* __restrict__ out, int E, int Nn) {
  const int wave = threadIdx.x >> 5;
  const int lane = threadIdx.x & 31;
  long long e = (long long)blockIdx.x * (blockDim.x >> 5) + wave;
  if (e >= (long long)E + Nn) return;
  int src, dst; float ew;
  if (e < E) {
    int ei = (int)e;
    src = edge_list[2 * ei]; dst = edge_list[2 * ei + 1];
    ew = edge_weight[ei];
  } else {
    src = dst = (int)(e - E);
    ew = 1.0f;
  }
  const int h = lane >> 2;  // lane handles dims d = 4*lane .. 4*lane+3, all head h
  float w = alpha_in[src * NUM_HEAD + h] + alpha_out[dst * NUM_HEAD + h];
  w = (w > 0.f) ? w : NEG_SLOPE * w;
  const float s = __expf(w - seg_max[dst * NUM_HEAD + h]) * ew;

  if ((lane & 3) == 0) atomicAdd(&norm_sum[dst * NUM_HEAD + h], s);

  v4f hv = *(const v4f*)(hidden + (size_t)src * OUT_DIM + lane * 4);
  float* op = out + (size_t)dst * OUT_DIM + lane * 4;
  atomicAdd(op + 0, s * hv.x);
  atomicAdd(op + 1, s * hv.y);
  atomicAdd(op + 2, s * hv.z);
  atomicAdd(op + 3, s * hv.w);
}

// ---------------------------------------------------------------------------
// K5: out = relu( accum / (norm_sum + counts*eps) )
//  (folds both scatter_mean divisions: accum/(counts*(norm_sum/counts+eps)))
// ---------------------------------------------------------------------------
__global__ void finalize_kernel(float* __restrict__ out,
                                const float* __restrict__ norm_sum,
                                const float* __restrict__ counts, int Nn) {
  int t = blockIdx.x * blockDim.x + threadIdx.x;
  if (t >= Nn * OUT_DIM) return;
  const int n = t >> 7;
  const int h = (t >> 4) & 7;
  float v = out[t] / (norm_sum[n * NUM_HEAD + h] + counts[n] * EPS);
  out[t] = fmaxf(v, 0.0f);
}

// ---------------------------------------------------------------------------
extern "C" void kernel_launch(void* const* d_in, const int* in_sizes, int n_in,
                              void* d_out, int out_size, void* d_ws, size_t ws_size,
                              hipStream_t stream) {
  const float* x           = (const float*)d_in[0];
  const int*   edge_list   = (const int*)d_in[1];
  const float* edge_weight = (const float*)d_in[2];
  const float* W           = (const float*)d_in[3];
  const float* b           = (const float*)d_in[4];
  const float* query       = (const float*)d_in[5];
  float* out = (float*)d_out;

  const int Nn = in_sizes[0] / IN_DIM;  // 100000 (multiple of 16)
  const int E  = in_sizes[1] / 2;       // 1600000

  // workspace layout (floats): hidden | alpha_in | alpha_out | seg_max | norm_sum | counts
  float* ws        = (float*)d_ws;
  float* hidden    = ws;
  float* alpha_in  = hidden    + (size_t)Nn * OUT_DIM;
  float* alpha_out = alpha_in  + (size_t)Nn * NUM_HEAD;
  float* seg_max   = alpha_out + (size_t)Nn * NUM_HEAD;
  float* norm_sum  = seg_max   + (size_t)Nn * NUM_HEAD;
  float* counts    = norm_sum  + (size_t)Nn * NUM_HEAD;

  {
    int total = Nn * OUT_DIM;
    init_kernel<<<(total + 255) / 256, 256, 0, stream>>>(out, seg_max, norm_sum,
                                                         counts, Nn);
  }

  gemm_wmma_kernel<<<Nn / 16, 256, 0, stream>>>(x, W, b, hidden);

  {
    int total = Nn * NUM_HEAD;
    alpha_kernel<<<(total + 255) / 256, 256, 0, stream>>>(hidden, query,
                                                          alpha_in, alpha_out, Nn);
  }

  {
    long long total = (long long)(E + Nn) * NUM_HEAD;
    attn_max_kernel<<<(unsigned)((total + 255) / 256), 256, 0, stream>>>(
        edge_list, alpha_in, alpha_out, seg_max, counts, E, Nn);
  }

  {
    long long edges = (long long)E + Nn;
    int edges_per_block = 256 / 32;
    scatter_kernel<<<(unsigned)((edges + edges_per_block - 1) / edges_per_block),
                     256, 0, stream>>>(edge_list, edge_weight, hidden, alpha_in,
                                       alpha_out, seg_max, norm_sum, out, E, Nn);
  }

  finalize_kernel<<<(Nn * OUT_DIM + 255) / 256, 256, 0, stream>>>(out, norm_sum,
                                                                  counts, Nn);
}